// DiffusionInteractionBlock_25082609009191
// MI455X (gfx1250) — compile-verified
//
#include <hip/hip_runtime.h>

#define NN 20000      // nodes
#define NE 320000     // edges
#define INV_SQRT3 0.5773502691896258f

typedef __attribute__((ext_vector_type(2))) float v2f;
typedef __attribute__((ext_vector_type(8))) float v8f;

__device__ __forceinline__ v8f wmma4(v2f a, v2f b, v8f c) {
  // V_WMMA_F32_16X16X4_F32 : D = A(16x4) * B(4x16) + C(16x16), fp32
  return __builtin_amdgcn_wmma_f32_16x16x4_f32(false, a, false, b, (short)0, c,
                                               false, false);
}

__device__ __forceinline__ float silu_f(float x) {
  return x * (1.0f / (1.0f + __expf(-x)));
}

__device__ __forceinline__ v8f zero8() {
  v8f z = {0.f, 0.f, 0.f, 0.f, 0.f, 0.f, 0.f, 0.f};
  return z;
}

// ---------------------------------------------------------------- zero scratch
__global__ __launch_bounds__(256) void zero_kernel(float4* p, long n4) {
  long i = (long)blockIdx.x * blockDim.x + threadIdx.x;
  if (i < n4) p[i] = make_float4(0.f, 0.f, 0.f, 0.f);
}

// ---------------------------------------------------- node transforms (WMMA)
// ns  = s @ Wsc           (N x 64)
// sup = s @ Wup0          (N x 64)
// vup[i] = v[:,:,i] @ Wup1 (3 planes, N x 64)
__global__ __launch_bounds__(256) void node_up_kernel(
    const float* __restrict__ node_feats, const float* __restrict__ Wsc,
    const float* __restrict__ Wup0, const float* __restrict__ Wup1,
    float* __restrict__ ns, float* __restrict__ sup, float* __restrict__ vup) {
  const int lane = threadIdx.x & 31;
  const int wv = threadIdx.x >> 5;
  const int tile = blockIdx.x * 8 + wv;
  if (tile >= NN / 16) return;
  const int n0 = tile * 16;
  const int row = lane & 15;
  const int hi = lane >> 4;
  const int col = row;
  const float* arow = node_feats + (long)(n0 + row) * 256;

  // ---- s @ Wsc and s @ Wup0 (shared A fragments)
  v8f accA[4], accB[4];
#pragma unroll
  for (int nt = 0; nt < 4; nt++) { accA[nt] = zero8(); accB[nt] = zero8(); }
#pragma unroll
  for (int kk = 0; kk < 16; kk++) {
    const int kb = kk * 4 + 2 * hi;
    v2f a = *(const v2f*)(arow + kb);
#pragma unroll
    for (int nt = 0; nt < 4; nt++) {
      v2f b0, b1;
      b0.x = Wsc[kb * 64 + nt * 16 + col];
      b0.y = Wsc[(kb + 1) * 64 + nt * 16 + col];
      accA[nt] = wmma4(a, b0, accA[nt]);
      b1.x = Wup0[kb * 64 + nt * 16 + col];
      b1.y = Wup0[(kb + 1) * 64 + nt * 16 + col];
      accB[nt] = wmma4(a, b1, accB[nt]);
    }
  }
#pragma unroll
  for (int nt = 0; nt < 4; nt++)
#pragma unroll
    for (int r = 0; r < 8; r++) {
      const int n = n0 + r + 8 * hi;
      ns[(long)n * 64 + nt * 16 + col] = accA[nt][r];
      sup[(long)n * 64 + nt * 16 + col] = accB[nt][r];
    }

  // ---- v planes: v[:,c,i] = node_feats[n, 64 + c*3 + i]
#pragma unroll
  for (int i = 0; i < 3; i++) {
    v8f acc[4];
#pragma unroll
    for (int nt = 0; nt < 4; nt++) acc[nt] = zero8();
#pragma unroll
    for (int kk = 0; kk < 16; kk++) {
      const int kb = kk * 4 + 2 * hi;
      v2f a;
      a.x = arow[64 + kb * 3 + i];
      a.y = arow[64 + (kb + 1) * 3 + i];
#pragma unroll
      for (int nt = 0; nt < 4; nt++) {
        v2f b;
        b.x = Wup1[kb * 64 + nt * 16 + col];
        b.y = Wup1[(kb + 1) * 64 + nt * 16 + col];
        acc[nt] = wmma4(a, b, acc[nt]);
      }
    }
#pragma unroll
    for (int nt = 0; nt < 4; nt++)
#pragma unroll
      for (int r = 0; r < 8; r++) {
        const int n = n0 + r + 8 * hi;
        vup[(long)i * NN * 64 + (long)n * 64 + nt * 16 + col] = acc[nt][r];
      }
  }
}

// ------------------------------------ fused edge MLP + tensor product + scatter
// Each wave owns TWO 16-edge M-tiles (32 edges) so every B (weight) fragment
// feeds two WMMAs: per k-step 2 A-loads + 8 B-loads -> 8 v_wmma.
__global__ __launch_bounds__(128) void edge_kernel(
    const float* __restrict__ edge_attrs, const float* __restrict__ edge_feats,
    const float* __restrict__ lengths, const int* __restrict__ edge_index,
    const float* __restrict__ W1, const float* __restrict__ b1,
    const float* __restrict__ W2, const float* __restrict__ b2,
    const float* __restrict__ W3, const float* __restrict__ ns,
    const float* __restrict__ sup, const float* __restrict__ vup,
    float* __restrict__ m0, float* __restrict__ m1) {
  __shared__ float lds_h[4][32 * 64];   // per-wave 32x64 tile (h1 then h2)
  __shared__ float lds_meta[4][32 * 8]; // per-edge: snd,rcv,y0,y1x,y1y,y1z
  const int lane = threadIdx.x & 31;
  const int wv = threadIdx.x >> 5;
  const int dt = blockIdx.x * 4 + wv;   // dual-tile id; 2500 blocks * 4 = E/32
  const int e0 = dt * 32;
  const int row = lane & 15;
  const int hi = lane >> 4;
  const int col = row;

  const int ea = e0 + row;              // lane's row in tile 0
  const int eb = e0 + 16 + row;         // lane's row in tile 1
  const int snd0 = edge_index[ea], rcv0 = edge_index[NE + ea];
  const int snd1 = edge_index[eb], rcv1 = edge_index[NE + eb];
  {  // every lane covers one of the 32 edges for the meta table
    const int mr = row + 16 * hi;
    const int me = e0 + mr;
    lds_meta[wv][mr * 8 + 0] = __int_as_float(hi ? snd1 : snd0);
    lds_meta[wv][mr * 8 + 1] = __int_as_float(hi ? rcv1 : rcv0);
    lds_meta[wv][mr * 8 + 2] = edge_attrs[me * 4 + 0];
    lds_meta[wv][mr * 8 + 3] = edge_attrs[me * 4 + 1];
    lds_meta[wv][mr * 8 + 4] = edge_attrs[me * 4 + 2];
    lds_meta[wv][mr * 8 + 5] = edge_attrs[me * 4 + 3];
  }
  const float* sp0 = ns + (long)snd0 * 64;
  const float* rp0 = ns + (long)rcv0 * 64;
  const float* sp1 = ns + (long)snd1 * 64;
  const float* rp1 = ns + (long)rcv1 * 64;
  const float* ep0 = edge_feats + (long)ea * 8;
  const float* ep1 = edge_feats + (long)eb * 8;
  const float len0 = lengths[ea];
  const float len1 = lengths[eb];

  // ======== phase 1: h1 = silu([ns[snd], ns[rcv], ef, len] @ W1 + b1), K=137
  v8f acc0[4], acc1[4];
#pragma unroll
  for (int nt = 0; nt < 4; nt++) { acc0[nt] = zero8(); acc1[nt] = zero8(); }
#pragma unroll 4
  for (int kk = 0; kk < 16; kk++) {  // K rows 0..63 from ns[snd]
    const int kb = kk * 4 + 2 * hi;
    v2f a0 = *(const v2f*)(sp0 + kb);
    v2f a1 = *(const v2f*)(sp1 + kb);
#pragma unroll
    for (int nt = 0; nt < 4; nt++) {
      v2f b;
      b.x = W1[kb * 64 + nt * 16 + col];
      b.y = W1[(kb + 1) * 64 + nt * 16 + col];
      acc0[nt] = wmma4(a0, b, acc0[nt]);
      acc1[nt] = wmma4(a1, b, acc1[nt]);
    }
  }
#pragma unroll 4
  for (int kk = 0; kk < 16; kk++) {  // K rows 64..127 from ns[rcv]
    const int kb = kk * 4 + 2 * hi;
    v2f a0 = *(const v2f*)(rp0 + kb);
    v2f a1 = *(const v2f*)(rp1 + kb);
#pragma unroll
    for (int nt = 0; nt < 4; nt++) {
      v2f b;
      b.x = W1[(64 + kb) * 64 + nt * 16 + col];
      b.y = W1[(64 + kb + 1) * 64 + nt * 16 + col];
      acc0[nt] = wmma4(a0, b, acc0[nt]);
      acc1[nt] = wmma4(a1, b, acc1[nt]);
    }
  }
#pragma unroll
  for (int kk = 0; kk < 2; kk++) {   // K rows 128..135 from edge_feats
    const int kb = kk * 4 + 2 * hi;
    v2f a0 = *(const v2f*)(ep0 + kb);
    v2f a1 = *(const v2f*)(ep1 + kb);
#pragma unroll
    for (int nt = 0; nt < 4; nt++) {
      v2f b;
      b.x = W1[(128 + kb) * 64 + nt * 16 + col];
      b.y = W1[(128 + kb + 1) * 64 + nt * 16 + col];
      acc0[nt] = wmma4(a0, b, acc0[nt]);
      acc1[nt] = wmma4(a1, b, acc1[nt]);
    }
  }
  {                                   // K row 136 = lengths; rows 137..139 = 0
    v2f a0, a1;
    a0.x = hi ? 0.f : len0;  a0.y = 0.f;
    a1.x = hi ? 0.f : len1;  a1.y = 0.f;
#pragma unroll
    for (int nt = 0; nt < 4; nt++) {
      v2f b;
      b.x = hi ? 0.f : W1[136 * 64 + nt * 16 + col];
      b.y = 0.f;
      acc0[nt] = wmma4(a0, b, acc0[nt]);
      acc1[nt] = wmma4(a1, b, acc1[nt]);
    }
  }
#pragma unroll
  for (int nt = 0; nt < 4; nt++) {   // bias + silu -> LDS
    const float bias = b1[nt * 16 + col];
#pragma unroll
    for (int r = 0; r < 8; r++) {
      lds_h[wv][(r + 8 * hi) * 64 + nt * 16 + col] = silu_f(acc0[nt][r] + bias);
      lds_h[wv][(16 + r + 8 * hi) * 64 + nt * 16 + col] =
          silu_f(acc1[nt][r] + bias);
    }
  }

  // ======== phase 2: h2 = silu(h1 @ W2 + b2), K=64
#pragma unroll
  for (int nt = 0; nt < 4; nt++) { acc0[nt] = zero8(); acc1[nt] = zero8(); }
#pragma unroll 4
  for (int kk = 0; kk < 16; kk++) {
    const int kb = kk * 4 + 2 * hi;
    v2f a0 = *(const v2f*)&lds_h[wv][row * 64 + kb];
    v2f a1 = *(const v2f*)&lds_h[wv][(16 + row) * 64 + kb];
#pragma unroll
    for (int nt = 0; nt < 4; nt++) {
      v2f b;
      b.x = W2[kb * 64 + nt * 16 + col];
      b.y = W2[(kb + 1) * 64 + nt * 16 + col];
      acc0[nt] = wmma4(a0, b, acc0[nt]);
      acc1[nt] = wmma4(a1, b, acc1[nt]);
    }
  }
  float h2a[4][8], h2b[4][8];
#pragma unroll
  for (int nt = 0; nt < 4; nt++) {
    const float bias = b2[nt * 16 + col];
#pragma unroll
    for (int r = 0; r < 8; r++) {
      h2a[nt][r] = silu_f(acc0[nt][r] + bias);
      h2b[nt][r] = silu_f(acc1[nt][r] + bias);
    }
  }
#pragma unroll
  for (int nt = 0; nt < 4; nt++)
#pragma unroll
    for (int r = 0; r < 8; r++) {
      lds_h[wv][(r + 8 * hi) * 64 + nt * 16 + col] = h2a[nt][r];
      lds_h[wv][(16 + r + 8 * hi) * 64 + nt * 16 + col] = h2b[nt][r];
    }

  // ======== phase 3: tw = h2 @ W3 in 4 blocks w0..w3, each fused with its
  //          tensor product and atomically scattered by rcv
#pragma unroll
  for (int g = 0; g < 4; g++) {
    v8f t0[4], t1[4];
#pragma unroll
    for (int nt = 0; nt < 4; nt++) { t0[nt] = zero8(); t1[nt] = zero8(); }
#pragma unroll 4
    for (int kk = 0; kk < 16; kk++) {
      const int kb = kk * 4 + 2 * hi;
      v2f a0 = *(const v2f*)&lds_h[wv][row * 64 + kb];
      v2f a1 = *(const v2f*)&lds_h[wv][(16 + row) * 64 + kb];
#pragma unroll
      for (int nt = 0; nt < 4; nt++) {
        v2f b;
        b.x = W3[kb * 256 + g * 64 + nt * 16 + col];
        b.y = W3[(kb + 1) * 256 + g * 64 + nt * 16 + col];
        t0[nt] = wmma4(a0, b, t0[nt]);
        t1[nt] = wmma4(a1, b, t1[nt]);
      }
    }
#pragma unroll
    for (int t = 0; t < 2; t++) {
#pragma unroll
      for (int nt = 0; nt < 4; nt++) {
        const int c = nt * 16 + col;
#pragma unroll
        for (int r = 0; r < 8; r++) {
          const int rr = t * 16 + r + 8 * hi;  // edge row within the 32
          const int s_e = __float_as_int(lds_meta[wv][rr * 8 + 0]);
          const int r_e = __float_as_int(lds_meta[wv][rr * 8 + 1]);
          const float y0 = lds_meta[wv][rr * 8 + 2];
          const float wgt = t ? t1[nt][r] : t0[nt][r];
          if (g == 0) {  // p0 = xs*y0*w0 -> m0[:, 0:64]
            const float val = sup[(long)s_e * 64 + c] * y0 * wgt;
            unsafeAtomicAdd(&m0[(long)r_e * 128 + c], val);
          } else if (g == 1) {  // p1 = (xv.y1)*inv_sqrt3*w1 -> m0[:, 64:128]
            const float dot =
                vup[(long)s_e * 64 + c] * lds_meta[wv][rr * 8 + 3] +
                vup[(long)NN * 64 + (long)s_e * 64 + c] *
                    lds_meta[wv][rr * 8 + 4] +
                vup[(long)2 * NN * 64 + (long)s_e * 64 + c] *
                    lds_meta[wv][rr * 8 + 5];
            unsafeAtomicAdd(&m0[(long)r_e * 128 + 64 + c],
                            dot * INV_SQRT3 * wgt);
          } else if (g == 2) {  // p2 = xs*y1_i*w2 -> m1[i][:, 0:64]
            const float xs = sup[(long)s_e * 64 + c];
#pragma unroll
            for (int i = 0; i < 3; i++)
              unsafeAtomicAdd(&m1[(long)i * NN * 128 + (long)r_e * 128 + c],
                              xs * lds_meta[wv][rr * 8 + 3 + i] * wgt);
          } else {  // p3 = xv_i*y0*w3 -> m1[i][:, 64:128]
#pragma unroll
            for (int i = 0; i < 3; i++)
              unsafeAtomicAdd(
                  &m1[(long)i * NN * 128 + (long)r_e * 128 + 64 + c],
                  vup[(long)i * NN * 64 + (long)s_e * 64 + c] * y0 * wgt);
          }
        }
      }
    }
  }
}

// -------------------------------------------------- output GEMMs + interleave
__global__ __launch_bounds__(256) void out_kernel(
    const float* __restrict__ m0, const float* __restrict__ m1,
    const float* __restrict__ Ws, const float* __restrict__ Wv,
    float* __restrict__ out) {
  const int lane = threadIdx.x & 31;
  const int wv = threadIdx.x >> 5;
  const int tile = blockIdx.x * 8 + wv;
  if (tile >= NN / 16) return;
  const int n0 = tile * 16;
  const int row = lane & 15;
  const int hi = lane >> 4;
  const int col = row;

  v8f acc[4][4];  // [j][nt], j=0 scalar, j=1..3 vector planes
#pragma unroll
  for (int j = 0; j < 4; j++)
#pragma unroll
    for (int nt = 0; nt < 4; nt++) acc[j][nt] = zero8();

#pragma unroll 4
  for (int kk = 0; kk < 32; kk++) {  // out_s = m0 @ Ws, K=128
    const int kb = kk * 4 + 2 * hi;
    v2f a = *(const v2f*)(m0 + (long)(n0 + row) * 128 + kb);
#pragma unroll
    for (int nt = 0; nt < 4; nt++) {
      v2f b;
      b.x = Ws[kb * 64 + nt * 16 + col];
      b.y = Ws[(kb + 1) * 64 + nt * 16 + col];
      acc[0][nt] = wmma4(a, b, acc[0][nt]);
    }
  }
#pragma unroll
  for (int i = 0; i < 3; i++) {      // out_v[i] = m1[i] @ Wv, K=128
#pragma unroll 4
    for (int kk = 0; kk < 32; kk++) {
      const int kb = kk * 4 + 2 * hi;
      v2f a =
          *(const v2f*)(m1 + (long)i * NN * 128 + (long)(n0 + row) * 128 + kb);
#pragma unroll
      for (int nt = 0; nt < 4; nt++) {
        v2f b;
        b.x = Wv[kb * 64 + nt * 16 + col];
        b.y = Wv[(kb + 1) * 64 + nt * 16 + col];
        acc[1 + i][nt] = wmma4(a, b, acc[1 + i][nt]);
      }
    }
  }
  const float inv = 1.0f / 16.0f;  // / AVG
#pragma unroll
  for (int nt = 0; nt < 4; nt++) {
    const int c = nt * 16 + col;
#pragma unroll
    for (int r = 0; r < 8; r++) {
      const int n = n0 + r + 8 * hi;
      float4 o = make_float4(acc[0][nt][r] * inv, acc[1][nt][r] * inv,
                             acc[2][nt][r] * inv, acc[3][nt][r] * inv);
      *(float4*)(out + (long)n * 256 + c * 4) = o;  // (N, 64, 4) interleaved
    }
  }
}

extern "C" void kernel_launch(void* const* d_in, const int* in_sizes, int n_in,
                              void* d_out, int out_size, void* d_ws,
                              size_t ws_size, hipStream_t stream) {
  (void)in_sizes; (void)n_in; (void)out_size; (void)ws_size;
  const float* node_feats = (const float*)d_in[0];
  const float* edge_attrs = (const float*)d_in[1];
  const float* edge_feats = (const float*)d_in[2];
  const float* lengths    = (const float*)d_in[3];
  const int*   edge_index = (const int*)d_in[4];
  const float* Wsc  = (const float*)d_in[5];
  const float* Wup0 = (const float*)d_in[6];
  const float* Wup1 = (const float*)d_in[7];
  const float* W1   = (const float*)d_in[8];
  const float* b1   = (const float*)d_in[9];
  const float* W2   = (const float*)d_in[10];
  const float* b2   = (const float*)d_in[11];
  const float* W3   = (const float*)d_in[12];
  const float* Ws   = (const float*)d_in[13];
  const float* Wv   = (const float*)d_in[14];
  float* out = (float*)d_out;

  float* ws  = (float*)d_ws;
  float* ns  = ws;                          // N*64
  float* sup = ns + (long)NN * 64;          // N*64
  float* vup = sup + (long)NN * 64;         // 3*N*64
  float* m0  = vup + (long)3 * NN * 64;     // N*128   (atomic accum)
  float* m1  = m0 + (long)NN * 128;         // 3*N*128 (atomic accum)

  const long zn4 = (long)NN * 512 / 4;      // zero m0+m1 (contiguous)
  zero_kernel<<<(int)((zn4 + 255) / 256), 256, 0, stream>>>((float4*)m0, zn4);

  node_up_kernel<<<(NN / 16 + 7) / 8, 256, 0, stream>>>(node_feats, Wsc, Wup0,
                                                        Wup1, ns, sup, vup);
  edge_kernel<<<NE / 32 / 4, 128, 0, stream>>>(edge_attrs, edge_feats, lengths,
                                               edge_index, W1, b1, W2, b2, W3,
                                               ns, sup, vup, m0, m1);
  out_kernel<<<(NN / 16 + 7) / 8, 256, 0, stream>>>(m0, m1, Ws, Wv, out);
}